// MultiHeadAttention_16698832847504
// MI455X (gfx1250) — compile-verified
//
#include <hip/hip_runtime.h>

// MI455X / gfx1250 fused multi-head attention (fp32, V_WMMA_F32_16X16X4_F32).
// Roofline: ~24 GFLOP vs ~1.07 GB of HBM traffic (attn_w output alone is
// 1 GiB) -> store-bound. Design: fp32 WMMA everywhere (precision-free since
// compute is not the bottleneck), attn_w written exactly once, normalized,
// as b128 stores; score tiles recomputed in a cheap second register pass.

typedef __attribute__((ext_vector_type(2))) float v2f;
typedef __attribute__((ext_vector_type(8))) float v8f;

#define BATCH   16
#define SEQ     1024
#define DMODEL  256
#define HEADS   16
#define DKDIM   16
#define WPB     4          // waves per block (wave32)
#define LDSW    20         // padded LDS row stride: 80 B = 16B-aligned, conflict-free

// ---------------------------------------------------------------------------
// Kernel 1: fused Q/K/V projection.  One wave computes one 16x16 output tile
// of X @ W (K=256 -> 64 WMMA_F32_16X16X4_F32 steps).  N-tile == one head, so
// results scatter straight into [B,H,S,DK] workspace layout.
// ---------------------------------------------------------------------------
__global__ __launch_bounds__(WPB * 32)
void proj_kernel(const float* __restrict__ Xq,
                 const float* __restrict__ Xk,
                 const float* __restrict__ Xv,
                 const float* __restrict__ Wq,
                 const float* __restrict__ bq,
                 const float* __restrict__ Wk,
                 const float* __restrict__ Wv,
                 const float* __restrict__ bv,
                 float* __restrict__ q_ws,
                 float* __restrict__ k_ws,
                 float* __restrict__ v_ws)
{
    const int lane = threadIdx.x & 31;
    const int wave = threadIdx.x >> 5;
    const int gid  = blockIdx.x * WPB + wave;

    const int mat = gid >> 14;            // 0=q, 1=k, 2=v  (16384 tiles each)
    int rem       = gid & 16383;
    const int b   = rem >> 10;            // batch
    rem          &= 1023;
    const int mt  = rem >> 4;             // row tile (0..63)
    const int nt  = rem & 15;             // head / N tile (0..15)
    const int m0  = mt * 16;
    const int n0  = nt * 16;

    const int lo = lane & 15;             // column (B/C layout) or row (A layout)
    const int hi = lane >> 4;             // lane half -> K sub-pair / row-half

    const float* X = (mat == 0) ? Xq : (mat == 1) ? Xk : Xv;
    const float* W = (mat == 0) ? Wq : (mat == 1) ? Wk : Wv;
    float*       O = (mat == 0) ? q_ws : (mat == 1) ? k_ws : v_ws;

    // A layout (16x4 f32): lanes 0-15 hold K={0,1}, lanes 16-31 hold K={2,3}.
    const float* xrow = X + ((size_t)b * SEQ + m0 + lo) * DMODEL;

    v8f c = {};
    #pragma unroll 4
    for (int k0 = 0; k0 < DMODEL; k0 += 4) {
        const int ka = k0 + 2 * hi;
        v2f a = *(const v2f*)(xrow + ka);                 // A: row=lo, dk pair
        v2f bb;                                           // B: rows ka,ka+1 col n0+lo
        bb.x = W[(size_t)ka       * DMODEL + n0 + lo];
        bb.y = W[(size_t)(ka + 1) * DMODEL + n0 + lo];
        c = __builtin_amdgcn_wmma_f32_16x16x4_f32(false, a, false, bb,
                                                  (short)0, c, false, false);
    }

    float bias = 0.0f;
    if (mat == 0)      bias = bq[n0 + lo];
    else if (mat == 2) bias = bv[n0 + lo];

    // C layout: VGPR r, lanes 0-15 -> row r, lanes 16-31 -> row r+8; col = lo.
    float* obase = O + ((size_t)b * HEADS + nt) * SEQ * DKDIM;
    #pragma unroll
    for (int r = 0; r < 8; ++r) {
        const int row = m0 + r + 8 * hi;
        obase[(size_t)row * DKDIM + lo] = c[r] + bias;
    }
}

// ---------------------------------------------------------------------------
// Kernel 2: attention.  One wave owns a (b,h, 16-row q strip).
// Pass 1: row sums of exp(q.k/4)*mask.  Pass 2: recompute tiles, normalize
// into a tiny LDS tile, emit attn_w as b128 stores, and feed the context
// WMMA from the same LDS tile (C-layout -> A-layout conversion).
// ---------------------------------------------------------------------------
__global__ __launch_bounds__(WPB * 32)
void attn_kernel(const float* __restrict__ q_ws,
                 const float* __restrict__ k_ws,
                 const float* __restrict__ v_ws,
                 const int*   __restrict__ mask,
                 float* __restrict__ out,    // [B,S,H*DK]
                 float* __restrict__ attn)   // [B,H,S,S]
{
    __shared__ float P_t[WPB][16][LDSW];    // 1.25 KB/wave, bank-conflict free

    const int lane = threadIdx.x & 31;
    const int wave = threadIdx.x >> 5;
    const int sid  = blockIdx.x * WPB + wave;

    const int b  = sid >> 10;               // / (HEADS * 64)
    const int h  = (sid >> 6) & 15;
    const int qt = sid & 63;
    const int m0 = qt * 16;
    const int lo = lane & 15;
    const int hi = lane >> 4;

    const size_t bh = (size_t)b * HEADS + h;
    const float* qp = q_ws + (bh * SEQ + m0) * DKDIM;
    const float* kp = k_ws + bh * SEQ * DKDIM;
    const float* vp = v_ws + bh * SEQ * DKDIM;
    const int*   mp = mask + b * SEQ;

    // q tile in A layout, kept in registers for both passes.
    v2f aq[4];
    #pragma unroll
    for (int c = 0; c < 4; ++c)
        aq[c] = *(const v2f*)(qp + (size_t)lo * DKDIM + 4 * c + 2 * hi);

    // exp(score/4)*mask tile for k-tile kt, in C layout.
    auto score_tile = [&](int kt) -> v8f {
        v8f s = {};
        const float* kbase = kp + (size_t)(kt * 16 + lo) * DKDIM;  // B col = key
        #pragma unroll
        for (int c = 0; c < 4; ++c) {
            v2f bb = *(const v2f*)(kbase + 4 * c + 2 * hi);        // K^T rows = dk
            s = __builtin_amdgcn_wmma_f32_16x16x4_f32(false, aq[c], false, bb,
                                                      (short)0, s, false, false);
        }
        const float mv = (float)mp[kt * 16 + lo];
        v8f e;
        #pragma unroll
        for (int r = 0; r < 8; ++r)
            e[r] = __expf(s[r] * 0.25f) * mv;
        return e;
    };

    // ---- Pass 1: row sums -------------------------------------------------
    float psum[8] = {};
    for (int kt = 0; kt < 64; ++kt) {
        v8f e = score_tile(kt);
        #pragma unroll
        for (int r = 0; r < 8; ++r) psum[r] += e[r];
    }
    float inv[8];
    #pragma unroll
    for (int r = 0; r < 8; ++r) {
        float s = psum[r];                  // reduce over N within 16-lane half
        s += __shfl_xor(s, 1, 32);
        s += __shfl_xor(s, 2, 32);
        s += __shfl_xor(s, 4, 32);
        s += __shfl_xor(s, 8, 32);
        inv[r] = 1.0f / (s + 1e-8f);        // row (r + 8*hi) of this strip
    }

    // ---- Pass 2: write attn_w (b128), accumulate context ------------------
    v8f ctx = {};
    float* attn_base = attn + (bh * SEQ + m0) * SEQ;
    const int srow = lane >> 1;             // row this lane stores (0..15)
    const int scol = (lane & 1) * 8;        // col base (0 or 8)

    for (int kt = 0; kt < 64; ++kt) {
        v8f e = score_tile(kt);
        // Normalize once into LDS: serves both the attn_w stream and the
        // context GEMM (so ctx accumulates already-normalized attn @ V).
        #pragma unroll
        for (int r = 0; r < 8; ++r)
            P_t[wave][r + 8 * hi][lo] = e[r] * inv[r];

        // attn_w tile as b128 stores (4x fewer store instrs on the 1 GiB stream)
        float4 t0 = *(const float4*)(&P_t[wave][srow][scol]);
        float4 t1 = *(const float4*)(&P_t[wave][srow][scol + 4]);
        float* dst = attn_base + (size_t)srow * SEQ + kt * 16 + scol;
        *(float4*)(dst)     = t0;
        *(float4*)(dst + 4) = t1;

        // context WMMA: A = normalized P tile (LDS, A layout), B = V tile
        const float* vbase = vp + (size_t)(kt * 16) * DKDIM;
        #pragma unroll
        for (int c = 0; c < 4; ++c) {
            v2f a = *(const v2f*)(&P_t[wave][lo][4 * c + 2 * hi]);
            v2f bb;
            bb.x = vbase[(size_t)(4 * c + 2 * hi)     * DKDIM + lo];
            bb.y = vbase[(size_t)(4 * c + 2 * hi + 1) * DKDIM + lo];
            ctx = __builtin_amdgcn_wmma_f32_16x16x4_f32(false, a, false, bb,
                                                        (short)0, ctx, false, false);
        }
    }

    // output[b, m0+row, h*16 + lo] = ctx (already normalized)
    float* obase = out + ((size_t)b * SEQ + m0) * DMODEL + h * DKDIM;
    #pragma unroll
    for (int r = 0; r < 8; ++r) {
        const int row = r + 8 * hi;
        obase[(size_t)row * DMODEL + lo] = ctx[r];
    }
}

// ---------------------------------------------------------------------------
extern "C" void kernel_launch(void* const* d_in, const int* in_sizes, int n_in,
                              void* d_out, int out_size, void* d_ws, size_t ws_size,
                              hipStream_t stream)
{
    const float* Q    = (const float*)d_in[0];
    const float* K    = (const float*)d_in[1];
    const float* V    = (const float*)d_in[2];
    const int*   mask = (const int*)  d_in[3];
    const float* Wq   = (const float*)d_in[4];
    const float* bq   = (const float*)d_in[5];
    const float* Wk   = (const float*)d_in[6];
    const float* Wv   = (const float*)d_in[7];
    const float* bv   = (const float*)d_in[8];

    const size_t proj_elems = (size_t)BATCH * SEQ * DMODEL;   // 4.19M floats each
    float* q_ws = (float*)d_ws;
    float* k_ws = q_ws + proj_elems;
    float* v_ws = k_ws + proj_elems;

    float* out  = (float*)d_out;                              // [B,S,256]
    float* attn = out + proj_elems;                           // [B,H,S,S]

    // 3 * 16 * 64 * 16 = 49152 projection tiles, 4 waves/block
    const int proj_waves = 3 * BATCH * (SEQ / 16) * (DMODEL / 16);
    proj_kernel<<<proj_waves / WPB, WPB * 32, 0, stream>>>(
        Q, K, V, Wq, bq, Wk, Wv, bv, q_ws, k_ws, v_ws);

    // 16 * 16 * 64 = 16384 attention strips, 4 waves/block
    const int attn_waves = BATCH * HEADS * (SEQ / 16);
    attn_kernel<<<attn_waves / WPB, WPB * 32, 0, stream>>>(
        q_ws, k_ws, v_ws, mask, out, attn);
}